// MultiBoxLoss_38895223833087
// MI455X (gfx1250) — compile-verified
//
#include <hip/hip_runtime.h>
#include <math.h>

#define NIMG 32
#define NPRI 8732
#define NCLS 81
#define NOBJ 24
#define THRESH 0.5f
#define NEG_RATIO 3
#define SORTN 16384
#define WPB 4                      // waves per block in perprior kernel

typedef __attribute__((ext_vector_type(2))) float v2f;
typedef __attribute__((ext_vector_type(8))) float v8f;
typedef __attribute__((ext_vector_type(4))) unsigned int u32x4;
typedef __attribute__((ext_vector_type(8))) int i32x8;
typedef __attribute__((ext_vector_type(4))) int i32x4;

// ---------------- init ----------------
__global__ void init_kernel(double* acc, int* npos_img, int* npos_tot) {
    int t = threadIdx.x;
    if (t < 4) acc[t] = 0.0;
    if (t < NIMG) npos_img[t] = 0;
    if (t == 0) *npos_tot = 0;
}

// ---------------- GT <-> prior matching (one block per image) ----------------
__global__ void match_kernel(const float* __restrict__ boxes, const int* __restrict__ labels,
                             const float* __restrict__ priors,
                             int* __restrict__ true_class, int* __restrict__ matched_obj,
                             float* __restrict__ best_iou_buf) {
    const int img = blockIdx.x;
    const int tid = threadIdx.x;
    __shared__ float gb[NOBJ][4];
    __shared__ float garea[NOBJ];
    __shared__ int   glab[NOBJ];
    __shared__ unsigned long long gkey[NOBJ];

    if (tid < NOBJ) {
        const float* b = boxes + ((size_t)img * NOBJ + tid) * 4;
        gb[tid][0] = b[0]; gb[tid][1] = b[1]; gb[tid][2] = b[2]; gb[tid][3] = b[3];
        garea[tid] = (b[2] - b[0]) * (b[3] - b[1]);
        glab[tid]  = labels[img * NOBJ + tid];
        gkey[tid]  = 0ull;
    }
    __syncthreads();

    unsigned long long lkey[NOBJ];
#pragma unroll
    for (int o = 0; o < NOBJ; ++o) lkey[o] = 0ull;

    for (int p = tid; p < NPRI; p += blockDim.x) {
        float cx = priors[p * 4 + 0], cy = priors[p * 4 + 1];
        float w  = priors[p * 4 + 2], h  = priors[p * 4 + 3];
        float x0 = cx - 0.5f * w, y0 = cy - 0.5f * h;
        float x1 = cx + 0.5f * w, y1 = cy + 0.5f * h;
        float pa = (x1 - x0) * (y1 - y0);
        float biou = -1.0f; int bobj = 0;
#pragma unroll
        for (int o = 0; o < NOBJ; ++o) {
            float ix0 = fmaxf(x0, gb[o][0]), iy0 = fmaxf(y0, gb[o][1]);
            float ix1 = fminf(x1, gb[o][2]), iy1 = fminf(y1, gb[o][3]);
            float iw = fmaxf(ix1 - ix0, 0.0f), ih = fmaxf(iy1 - iy0, 0.0f);
            float inter = iw * ih;
            float iou = inter / (pa + garea[o] - inter);
            if (iou > biou) { biou = iou; bobj = o; }   // first-index tie-break (argmax axis=0)
            unsigned long long key =
                ((unsigned long long)__float_as_uint(iou) << 32) |
                (unsigned long long)(0xFFFFFFFFu - (unsigned)p); // lowest-prior tie-break
            lkey[o] = (key > lkey[o]) ? key : lkey[o];
        }
        matched_obj[(size_t)img * NPRI + p]  = bobj;
        best_iou_buf[(size_t)img * NPRI + p] = biou;
    }
#pragma unroll
    for (int o = 0; o < NOBJ; ++o) atomicMax(&gkey[o], lkey[o]);
    __syncthreads();

    if (tid == 0) {  // sequential: duplicate best-priors -> last object wins (scatter order)
        for (int o = 0; o < NOBJ; ++o) {
            unsigned pr = 0xFFFFFFFFu - (unsigned)(gkey[o] & 0xFFFFFFFFull);
            matched_obj[(size_t)img * NPRI + pr]  = o;
            best_iou_buf[(size_t)img * NPRI + pr] = 1.0f;
        }
    }
    __syncthreads();

    for (int p = tid; p < NPRI; p += blockDim.x) {
        size_t idx = (size_t)img * NPRI + p;
        int o = matched_obj[idx];
        float iou = best_iou_buf[idx];
        true_class[idx] = (iou < THRESH) ? 0 : glab[o];
    }
}

// -------- per-prior: TDM double-buffered staging + WMMA log-softmax + CIoU --------
__global__ void __launch_bounds__(32 * WPB)
perprior_kernel(const float* __restrict__ locs, const float* __restrict__ scores,
                const float* __restrict__ boxes, const float* __restrict__ priors,
                const int* __restrict__ true_class, const int* __restrict__ matched_obj,
                float* __restrict__ conf_neg, double* __restrict__ accum,
                int* __restrict__ npos_img, int* __restrict__ npos_tot) {
    __shared__ float stage[WPB][2][16 * NCLS];          // 4*2*1296*4B = 41472 B

    const int lane   = threadIdx.x & 31;
    const int wave   = threadIdx.x >> 5;
    const int gwave  = blockIdx.x * WPB + wave;
    const int nwaves = gridDim.x * WPB;
    const int tiles_per_img = (NPRI + 15) / 16;          // 546
    const int total_tiles   = NIMG * tiles_per_img;

    // Issue a TDM 2D tile load: 16 rows x 81 f32, rows beyond image end read as zero.
    auto issue = [&](int tile, int parity) {
        const int img = tile / tiles_per_img;
        const int p0  = (tile % tiles_per_img) * 16;
        int valid = NPRI - p0; if (valid > 16) valid = 16;
        unsigned long long ga =
            (unsigned long long)(size_t)(scores + ((size_t)img * NPRI + p0) * NCLS);
        unsigned int la = (unsigned int)(size_t)(&stage[wave][parity][0]); // LDS offset (addr[31:0])

        u32x4 g0;
        g0.x = 1u;                                            // count=1, user descriptor
        g0.y = la;                                            // lds_addr [63:32]
        g0.z = (unsigned int)(ga & 0xFFFFFFFFull);            // global_addr [95:64]
        g0.w = (unsigned int)((ga >> 32) & 0x01FFFFFFull)     // global_addr [120:96]
             | (2u << 30);                                    // type=2 ("image")
        i32x8 g1;
        g1[0] = (2 << 16);          // workgroup_mask=0, data_size=2 (4B), no pad/iterate
        g1[1] = (NCLS << 16);       // tensor_dim0 = 81  (bits 79:48 low half)
        g1[2] = (valid << 16);      // tensor_dim0 hi=0, tensor_dim1 = valid rows
        g1[3] = (NCLS << 16);       // tensor_dim1 hi=0, tile_dim0 = 81
        g1[4] = 16;                 // tile_dim1 = 16, tile_dim2 = 0
        g1[5] = NCLS;               // tensor_dim0_stride = 81 (48-bit, low word)
        g1[6] = 0;                  // stride hi, tensor_dim1_stride lo
        g1[7] = 0;
        i32x4 z4 = {0, 0, 0, 0};                 // groups 2/3 unused (2-D tensor)
        i32x8 z8 = {0, 0, 0, 0, 0, 0, 0, 0};     // extra group (6-arg toolchain form)
        __builtin_amdgcn_tensor_load_to_lds(g0, g1, z4, z4, z8, 0);
    };

    if (gwave < total_tiles) issue(gwave, 0);

    int pi = 0;
    for (int tile = gwave; tile < total_tiles; tile += nwaves, pi ^= 1) {
        const int nxt = tile + nwaves;
        if (nxt < total_tiles) {
            issue(nxt, pi ^ 1);
            __builtin_amdgcn_s_wait_tensorcnt(1);  // in-order: <=1 outstanding => buffer pi ready
        } else {
            __builtin_amdgcn_s_wait_tensorcnt(0);
        }

        const int img = tile / tiles_per_img;
        const int p0  = (tile % tiles_per_img) * 16;

        // A-matrix layout (32-bit 16x4): lane<16 -> row=lane, K={0,1}; lane>=16 -> row=lane-16, K={2,3}
        const int koff = (lane >> 4) << 1;
        const float* rowp = &stage[wave][pi][(lane & 15) * NCLS];  // stride 81 words: bank-conflict-free

        float vals[42];
        float m = -3.0e38f;
#pragma unroll
        for (int it = 0; it < 21; ++it) {       // 21*4 = 84 >= 81 classes, pad with -big
            int c0 = it * 4 + koff;
            int c1 = c0 + 1;
            float a0 = rowp[(c0 < NCLS) ? c0 : 0];
            float a1 = rowp[(c1 < NCLS) ? c1 : 0];
            a0 = (c0 < NCLS) ? a0 : -3.0e38f;
            a1 = (c1 < NCLS) ? a1 : -3.0e38f;
            vals[2 * it]     = a0;
            vals[2 * it + 1] = a1;
            m = fmaxf(m, fmaxf(a0, a1));
        }
        // lanes l and l^16 hold complementary class halves of the same prior row
        float rowmax = fmaxf(m, __shfl_xor(m, 16, 32));

        // row-sums of exp on the matrix pipe: D += A x ones(4x16); exact f32 FMA accumulation
        v8f acc = {0.f, 0.f, 0.f, 0.f, 0.f, 0.f, 0.f, 0.f};
        v2f bone; bone.x = 1.0f; bone.y = 1.0f;
#pragma unroll
        for (int it = 0; it < 21; ++it) {
            v2f a;
            a.x = expf(vals[2 * it]     - rowmax);   // pads -> exp(-huge) = 0
            a.y = expf(vals[2 * it + 1] - rowmax);
            acc = __builtin_amdgcn_wmma_f32_16x16x4_f32(false, a, false, bone,
                                                        (short)0, acc, false, false);
        }

        // D layout: acc[v] @ lane<16 = rowsum(v); @ lane>=16 = rowsum(v+8); columns identical.
        int sel = lane & 7;
        float rs = acc[0];
        rs = (sel == 1) ? acc[1] : rs;
        rs = (sel == 2) ? acc[2] : rs;
        rs = (sel == 3) ? acc[3] : rs;
        rs = (sel == 4) ? acc[4] : rs;
        rs = (sel == 5) ? acc[5] : rs;
        rs = (sel == 6) ? acc[6] : rs;
        rs = (sel == 7) ? acc[7] : rs;

        const int q = (lane & 7) + ((lane & 16) ? 8 : 0);     // row this lane publishes
        const int srcl = (lane < 16) ? lane : (lane - 8);     // lane holding rowmax(q)
        float wmax = __shfl(rowmax, srcl, 32);

        const int p = p0 + q;
        const bool writer = ((lane & 8) == 0) && (p < NPRI);
        if (writer) {
            size_t idx = (size_t)img * NPRI + p;
            int cls = true_class[idx];
            float xc = stage[wave][pi][q * NCLS + cls];        // gather from LDS tile
            float conf = logf(rs) + wmax - xc;                 // -log_softmax[cls] >= 0
            bool pos = (cls != 0);
            conf_neg[idx] = pos ? 0.0f : conf;
            if (pos) {
                atomicAdd(&accum[1], (double)conf);
                atomicAdd(&npos_img[img], 1);
                atomicAdd(npos_tot, 1);

                // decode predicted box (SSD variances 10,5) then CIoU vs matched GT
                float g0 = locs[idx * 4 + 0], g1 = locs[idx * 4 + 1];
                float g2 = locs[idx * 4 + 2], g3 = locs[idx * 4 + 3];
                float pcx = priors[p * 4 + 0], pcy = priors[p * 4 + 1];
                float pw  = priors[p * 4 + 2], ph  = priors[p * 4 + 3];
                float cx = g0 * pw * 0.1f + pcx;
                float cy = g1 * ph * 0.1f + pcy;
                float w  = expf(g2 * 0.2f) * pw;
                float h  = expf(g3 * 0.2f) * ph;
                float b1x0 = cx - 0.5f * w, b1y0 = cy - 0.5f * h;
                float b1x1 = cx + 0.5f * w, b1y1 = cy + 0.5f * h;
                int o = matched_obj[idx];
                const float* gbp = boxes + ((size_t)img * NOBJ + o) * 4;
                float b2x0 = gbp[0], b2y0 = gbp[1], b2x1 = gbp[2], b2y1 = gbp[3];

                float w1 = b1x1 - b1x0, h1 = b1y1 - b1y0;
                float w2 = b2x1 - b2x0, h2 = b2y1 - b2y0;
                float area1 = w1 * h1, area2 = w2 * h2;
                float c1x = 0.5f * (b1x1 + b1x0), c1y = 0.5f * (b1y1 + b1y0);
                float c2x = 0.5f * (b2x1 + b2x0), c2y = 0.5f * (b2y1 + b2y0);
                float inw = fmaxf(fminf(b1x1, b2x1) - fmaxf(b1x0, b2x0), 0.0f);
                float inh = fmaxf(fminf(b1y1, b2y1) - fmaxf(b1y0, b2y0), 0.0f);
                float ow  = fmaxf(fmaxf(b1x1, b2x1) - fminf(b1x0, b2x0), 0.0f);
                float oh  = fmaxf(fmaxf(b1y1, b2y1) - fminf(b1y0, b2y0), 0.0f);
                float idg = (c2x - c1x) * (c2x - c1x) + (c2y - c1y) * (c2y - c1y);
                float odg = ow * ow + oh * oh;
                float inner = inw * inh;
                float iou = inner / (area1 + area2 - inner);
                float dat = atanf(w2 / h2) - atanf(w1 / h1);
                float v = 0.40528473456935108577f * dat * dat;   // 4/pi^2
                float alpha = v / (1.0f - iou + v);
                float cl = 1.0f - iou + idg / odg + alpha * v;
                atomicAdd(&accum[0], (double)cl);
            }
        }
    }
}

// ---------------- per-image hard-negative top-k via LDS bitonic sort ----------------
__global__ void topk_kernel(const float* __restrict__ conf_neg, const int* __restrict__ npos_img,
                            double* __restrict__ accum) {
    __shared__ float s[SORTN];
    __shared__ float red[256];
    const int img = blockIdx.x, tid = threadIdx.x, nt = blockDim.x;

    for (int i = tid; i < SORTN; i += nt)
        s[i] = (i < NPRI) ? conf_neg[(size_t)img * NPRI + i] : 0.0f;   // conf>=0, pad 0 is inert
    __syncthreads();

    for (int k = 2; k <= SORTN; k <<= 1) {
        for (int j = k >> 1; j > 0; j >>= 1) {
            for (int i = tid; i < SORTN; i += nt) {
                int ixj = i ^ j;
                if (ixj > i) {
                    bool desc = ((i & k) == 0);
                    float a = s[i], b = s[ixj];
                    if (desc ? (a < b) : (a > b)) { s[i] = b; s[ixj] = a; }
                }
            }
            __syncthreads();
        }
    }

    int nh = NEG_RATIO * npos_img[img];
    if (nh > SORTN) nh = SORTN;
    float part = 0.0f;
    for (int i = tid; i < nh; i += nt) part += s[i];
    red[tid] = part;
    __syncthreads();
    for (int st = nt >> 1; st > 0; st >>= 1) {
        if (tid < st) red[tid] += red[tid + st];
        __syncthreads();
    }
    if (tid == 0) atomicAdd(&accum[2], (double)red[0]);
}

// ---------------- finalize ----------------
__global__ void finalize_kernel(const double* __restrict__ accum, const int* __restrict__ npos_tot,
                                float* __restrict__ out) {
    double np = (double)(*npos_tot);
    double conf = (accum[2] + accum[1]) / np;
    double loc  = accum[0] / np;
    out[0] = (float)(conf + loc);
}

extern "C" void kernel_launch(void* const* d_in, const int* in_sizes, int n_in,
                              void* d_out, int out_size, void* d_ws, size_t ws_size,
                              hipStream_t stream) {
    const float* locs   = (const float*)d_in[0];
    const float* scores = (const float*)d_in[1];
    const float* boxes  = (const float*)d_in[2];
    const int*   labels = (const int*)d_in[3];
    const float* priors = (const float*)d_in[4];
    float* out = (float*)d_out;

    char* ws = (char*)d_ws;
    double* acc        = (double*)ws;                         // [0]=loc [1]=conf_pos [2]=hard_neg
    int*    npos_img   = (int*)(ws + 32);                     // 32
    int*    npos_tot   = (int*)(ws + 32 + NIMG * 4);          // 1
    int*    true_class = (int*)(ws + 256);                    // N*P
    int*    matched    = true_class + (size_t)NIMG * NPRI;    // N*P
    float*  conf_neg   = (float*)(matched + (size_t)NIMG * NPRI); // N*P (also best_iou temp)

    init_kernel<<<1, 64, 0, stream>>>(acc, npos_img, npos_tot);
    match_kernel<<<NIMG, 256, 0, stream>>>(boxes, labels, priors, true_class, matched, conf_neg);
    perprior_kernel<<<1024, 32 * WPB, 0, stream>>>(locs, scores, boxes, priors, true_class, matched,
                                                   conf_neg, acc, npos_img, npos_tot);
    topk_kernel<<<NIMG, 256, 0, stream>>>(conf_neg, npos_img, acc);
    finalize_kernel<<<1, 1, 0, stream>>>(acc, npos_tot, out);
}